// NegativeSampling_15960098472432
// MI455X (gfx1250) — compile-verified
//
#include <hip/hip_runtime.h>

typedef __attribute__((ext_vector_type(2))) float v2f;
typedef __attribute__((ext_vector_type(4))) float v4f;
typedef __attribute__((ext_vector_type(8))) float v8f;

constexpr int E = 128;            // embed size
constexpr int N_NEG = 16;         // negatives per token
constexpr int WAVES_PER_BLOCK = 8;

__device__ __forceinline__ float log_sigmoid(float x) {
    // stable: min(x,0) - log1p(exp(-|x|))
    return fminf(x, 0.0f) - log1pf(__expf(-fabsf(x)));
}

__global__ void negsamp_zero_out(float* out) { out[0] = 0.0f; }

__global__ __launch_bounds__(WAVES_PER_BLOCK * 32)
void negsamp_wmma_kernel(const int* __restrict__ sentence,   // [T]
                         const float* __restrict__ context,  // [T,E]
                         const int* __restrict__ neg,        // [T,N]
                         const float* __restrict__ W,        // [V,E]
                         float* __restrict__ out,            // [1]
                         int n_tokens)
{
    const int lane = threadIdx.x & 31;
    const int wave = threadIdx.x >> 5;
    const int tok  = blockIdx.x * WAVES_PER_BLOCK + wave;

    float wave_val = 0.0f;
    if (tok < n_tokens) {   // uniform per wave; grid is sized to full waves
        // ---- context slice for the positive dot: lane holds c[4*lane .. 4*lane+3]
        const float* crow0 = context + (size_t)tok * E;
        const v4f c = *(const v4f*)(crow0 + lane * 4);

        // A (16x4 f32) layout: lanes 0-15: row=lane,    K={4s+0,4s+1}
        //                      lanes 16-31: row=lane-16, K={4s+2,4s+3}
        const int r    = lane & 15;
        const int koff = (lane < 16) ? 0 : 2;
        const long long nidx = neg[(size_t)tok * N_NEG + r];
        const float* arow = W + (size_t)nidx * E + koff;

        // B (4x16 f32): only column N==0 non-zero. Lane 0 supplies K={0,1},
        // lane 16 supplies K={2,3} -> same koff addressing as A, sourced from
        // the context row in global memory (L0-cached; no LDS broadcast needed).
        const float* brow = crow0 + koff;
        const bool  is_n0 = ((lane & 15) == 0);   // lanes 0 and 16

        v8f acc = {0.f, 0.f, 0.f, 0.f, 0.f, 0.f, 0.f, 0.f};

        #pragma unroll 8
        for (int s = 0; s < E / 4; ++s) {
            v2f a    = *(const v2f*)(arow + 4 * s);
            v2f braw = *(const v2f*)(brow + 4 * s);
            v2f b;
            b.x = is_n0 ? braw.x : 0.0f;
            b.y = is_n0 ? braw.y : 0.0f;
            acc = __builtin_amdgcn_wmma_f32_16x16x4_f32(
                /*neg_a=*/false, a, /*neg_b=*/false, b,
                /*c_mod=*/(short)0, acc, /*reuse_a=*/false, /*reuse_b=*/false);
        }

        // neg scores sit in column N==0 of the 16x16 D tile:
        //   lane 0: acc[j] = score of negative j (M=0..7); lane 16: negative 8+j
        float nl = 0.0f;
        if (is_n0) {
            #pragma unroll
            for (int j = 0; j < 8; ++j) nl += log_sigmoid(-acc[j]);
        }

        // ---- positive dot: per-lane FMA + full butterfly reduce ----
        const long long pidx = sentence[tok];
        const v4f e = *(const v4f*)(W + (size_t)pidx * E + lane * 4);
        float pd = e.x * c.x + e.y * c.y + e.z * c.z + e.w * c.w;
        #pragma unroll
        for (int off = 16; off >= 1; off >>= 1) pd += __shfl_xor(pd, off, 32);
        float pl = (lane == 0) ? log_sigmoid(pd) : 0.0f;

        // ---- wave total: only lanes 0 and 16 hold non-zero contributions ----
        float v = nl + pl;
        v += __shfl_xor(v, 16, 32);   // lane0 now has nl0 + nl16 + pl
        wave_val = v;
    }

    __shared__ float smem[WAVES_PER_BLOCK];
    if (lane == 0) smem[wave] = wave_val;
    __syncthreads();
    if (threadIdx.x == 0) {
        float bs = 0.0f;
        #pragma unroll
        for (int i = 0; i < WAVES_PER_BLOCK; ++i) bs += smem[i];
        atomicAdd(out, -bs);   // reference returns -sum(pos_loss + neg_loss)
    }
}

extern "C" void kernel_launch(void* const* d_in, const int* in_sizes, int n_in,
                              void* d_out, int out_size, void* d_ws, size_t ws_size,
                              hipStream_t stream) {
    // setup_inputs order: sentence [B,S] int, context [B,S,E] f32,
    //                     neg_samples [B,S,N] int, W [V,E] f32
    const int*   sentence = (const int*)d_in[0];
    const float* context  = (const float*)d_in[1];
    const int*   neg      = (const int*)d_in[2];
    const float* W        = (const float*)d_in[3];
    float*       out      = (float*)d_out;

    const int n_tokens = in_sizes[0];  // B*S = 32768

    negsamp_zero_out<<<1, 1, 0, stream>>>(out);

    const int blocks = (n_tokens + WAVES_PER_BLOCK - 1) / WAVES_PER_BLOCK;
    negsamp_wmma_kernel<<<blocks, WAVES_PER_BLOCK * 32, 0, stream>>>(
        sentence, context, neg, W, out, n_tokens);
}